// GCN_89472758710435
// MI455X (gfx1250) — compile-verified
//
#include <hip/hip_runtime.h>
#include <math.h>

#define NN 100000
#define NE 1600000
#define DK 64       // input feature width for every GEMM (D_IN == D_HID == 64)

typedef __attribute__((ext_vector_type(2))) float v2f;
typedef __attribute__((ext_vector_type(8))) float v8f;

// -------------------- degree / norm --------------------
__global__ void deg_init(float* deg, int n) {
    int i = blockIdx.x * blockDim.x + threadIdx.x;
    if (i < n) deg[i] = 1.0f;                      // self-loop contributes 1
}

__global__ void deg_count(const int* __restrict__ dst, float* __restrict__ deg, int e) {
    int i = blockIdx.x * blockDim.x + threadIdx.x;
    if (i < e) atomicAdd(&deg[dst[i]], 1.0f);
}

__global__ void deg_rsqrt(float* deg, int n) {
    int i = blockIdx.x * blockDim.x + threadIdx.x;
    if (i < n) deg[i] = rsqrtf(deg[i]);            // deg >= 1 always
}

// -------------------- fused GEMM + self-loop epilogue --------------------
// One wave computes a full 16-row x NCOLS output block with NT accumulators.
// A: [nRows x 64] row-major (nRows % 16 == 0), W: [64 x NCOLS] row-major.
// Writes h = A@W and agg = h * dis^2 (self-loop message initializes the
// aggregation buffer; edges are atomically added later).
template <int NCOLS>
__global__ void gemm_wmma_fused(const float* __restrict__ A, const float* __restrict__ W,
                                const float* __restrict__ dis,
                                float* __restrict__ h, float* __restrict__ agg,
                                int nRowTiles) {
    constexpr int NT = (NCOLS + 15) / 16;
    const int lane = threadIdx.x & 31;
    const int warp = threadIdx.x >> 5;
    const int rt = blockIdx.x * (blockDim.x >> 5) + warp;
    if (rt >= nRowTiles) return;                   // wave-uniform exit

    const int mBase = rt << 4;
    const int l15  = lane & 15;
    const int kOff = (lane >> 4) << 1;             // lanes 16..31 hold K+2, K+3

    const float* __restrict__ arow = A + (mBase + l15) * DK + kOff;

    v8f zero = {};
    v8f acc[NT];
#pragma unroll
    for (int t = 0; t < NT; ++t) acc[t] = zero;

    // Per-lane, per-tile W base pointers; column clamped for the ragged tile so
    // loads stay in-bounds and EXEC stays all-ones at the WMMA.
    const float* wb[NT];
#pragma unroll
    for (int t = 0; t < NT; ++t) {
        int col = t * 16 + l15;
        if (col > NCOLS - 1) col = NCOLS - 1;
        wb[t] = W + kOff * NCOLS + col;
    }
    const float mask =
        (NCOLS % 16 == 0 || (NT - 1) * 16 + l15 < NCOLS) ? 1.0f : 0.0f;

#pragma unroll
    for (int k = 0; k < DK; k += 4) {
        v2f a;
        a.x = arow[k];
        a.y = arow[k + 1];
#pragma unroll
        for (int t = 0; t < NT; ++t) {
            v2f b;
            b.x = wb[t][k * NCOLS];                // compile-time immediate offsets
            b.y = wb[t][(k + 1) * NCOLS];
            if (NCOLS % 16 != 0 && t == NT - 1) { b.x *= mask; b.y *= mask; }
            acc[t] = __builtin_amdgcn_wmma_f32_16x16x4_f32(
                /*neg_a=*/false, a, /*neg_b=*/false, b,
                /*c_mod=*/(short)0, acc[t], /*reuse_a=*/false, /*reuse_b=*/false);
        }
    }

    const int rBase = mBase + ((lane >> 4) << 3);  // C row = v (lanes 0-15) / v+8
#pragma unroll
    for (int v = 0; v < 8; ++v) {
        const float s  = dis[rBase + v];
        const float s2 = s * s;
        float* __restrict__ hrow = h   + (rBase + v) * NCOLS;
        float* __restrict__ grow = agg + (rBase + v) * NCOLS;
#pragma unroll
        for (int t = 0; t < NT; ++t) {
            const int col = t * 16 + l15;
            if (NCOLS % 16 == 0 || col < NCOLS) {
                hrow[col] = acc[t][v];
                grow[col] = acc[t][v] * s2;        // self-loop message
            }
        }
    }
}

// -------------------- edge aggregation --------------------
// agg[dst, :] += h[src, :] * dis[src] * dis[dst]; float4 per thread.
template <int D>
__global__ void edge_msg(const float* __restrict__ h, const int* __restrict__ src,
                         const int* __restrict__ dst, const float* __restrict__ dis,
                         float* __restrict__ out, int total /* = E * D/4 */) {
    constexpr int Q = D / 4;
    int idx = blockIdx.x * blockDim.x + threadIdx.x;
    if (idx >= total) return;
    int e = idx / Q;
    int q = idx - e * Q;
    int s = src[e];
    int d = dst[e];
    float w = dis[s] * dis[d];
    const float4 hv = *(const float4*)(h + (size_t)s * D + q * 4);
    float* o = out + (size_t)d * D + q * 4;
    atomicAdd(o + 0, hv.x * w);
    atomicAdd(o + 1, hv.y * w);
    atomicAdd(o + 2, hv.z * w);
    atomicAdd(o + 3, hv.w * w);
}

template <int D, bool RELU>
__global__ void bias_act(float* __restrict__ out, const float* __restrict__ b, int total) {
    int idx = blockIdx.x * blockDim.x + threadIdx.x;
    if (idx < total) {
        int f = idx % D;
        float v = out[idx] + b[f];
        if (RELU) v = fmaxf(v, 0.0f);
        out[idx] = v;
    }
}

// -------------------- log_softmax over 40 features --------------------
__global__ void log_softmax40(const float* __restrict__ in, float* __restrict__ out, int n) {
    int i = blockIdx.x * blockDim.x + threadIdx.x;
    if (i >= n) return;
    const float* r = in + (size_t)i * 40;
    float m = -INFINITY;
#pragma unroll
    for (int j = 0; j < 40; ++j) m = fmaxf(m, r[j]);
    float s = 0.0f;
#pragma unroll
    for (int j = 0; j < 40; ++j) s += __expf(r[j] - m);
    float ls = __logf(s);
    float* o = out + (size_t)i * 40;
#pragma unroll
    for (int j = 0; j < 40; ++j) o[j] = r[j] - m - ls;
}

// -------------------- host entry --------------------
extern "C" void kernel_launch(void* const* d_in, const int* in_sizes, int n_in,
                              void* d_out, int out_size, void* d_ws, size_t ws_size,
                              hipStream_t stream) {
    const float* x  = (const float*)d_in[0];
    const int*   ei = (const int*)d_in[1];
    const float* W1 = (const float*)d_in[2];
    const float* b1 = (const float*)d_in[3];
    const float* W2 = (const float*)d_in[4];
    const float* b2 = (const float*)d_in[5];
    const float* W3 = (const float*)d_in[6];
    const float* b3 = (const float*)d_in[7];
    float* outp = (float*)d_out;

    const int N = NN, E = NE;
    const int* src = ei;          // edge_index[0]
    const int* dst = ei + E;      // edge_index[1]

    float* ws    = (float*)d_ws;
    float* dis   = ws;                        // N
    float* hbuf  = ws + N;                    // N*64  (GEMM output h)
    float* abufA = hbuf  + (size_t)N * 64;    // N*64  (aggregate ping)
    float* abufB = abufA + (size_t)N * 64;    // N*64  (aggregate pong)

    const int BT = 256;
    auto blocks = [](long long total, int bt) { return (int)((total + bt - 1) / bt); };

    // ---- symmetric-norm coefficients (shared by all layers) ----
    deg_init <<<blocks(N, BT), BT, 0, stream>>>(dis, N);
    deg_count<<<blocks(E, BT), BT, 0, stream>>>(dst, dis, E);
    deg_rsqrt<<<blocks(N, BT), BT, 0, stream>>>(dis, N);

    const int rowTiles = N >> 4;                 // 6250 (exact)
    const int gemmBlks = (rowTiles + 7) / 8;     // 8 waves / block

    // ---- layer 1: x -> abufA ----
    gemm_wmma_fused<64><<<gemmBlks, 256, 0, stream>>>(x, W1, dis, hbuf, abufA, rowTiles);
    edge_msg<64>      <<<blocks((long long)E * 16, BT), BT, 0, stream>>>(hbuf, src, dst, dis, abufA, E * 16);
    bias_act<64, true><<<blocks((long long)N * 64, BT), BT, 0, stream>>>(abufA, b1, N * 64);

    // ---- layer 2: abufA -> abufB (no aliasing of GEMM input) ----
    gemm_wmma_fused<64><<<gemmBlks, 256, 0, stream>>>(abufA, W2, dis, hbuf, abufB, rowTiles);
    edge_msg<64>      <<<blocks((long long)E * 16, BT), BT, 0, stream>>>(hbuf, src, dst, dis, abufB, E * 16);
    bias_act<64, true><<<blocks((long long)N * 64, BT), BT, 0, stream>>>(abufB, b2, N * 64);

    // ---- layer 3 (40 wide): abufB -> abufA[0 .. N*40) ----
    gemm_wmma_fused<40><<<gemmBlks, 256, 0, stream>>>(abufB, W3, dis, hbuf, abufA, rowTiles);
    edge_msg<40>       <<<blocks((long long)E * 10, BT), BT, 0, stream>>>(hbuf, src, dst, dis, abufA, E * 10);
    bias_act<40, false><<<blocks((long long)N * 40, BT), BT, 0, stream>>>(abufA, b3, N * 40);

    // ---- log_softmax -> d_out ----
    log_softmax40<<<blocks(N, BT), BT, 0, stream>>>(abufA, outp, N);
}